// NLF_12043088298490
// MI455X (gfx1250) — compile-verified
//
#include <hip/hip_runtime.h>
#include <cstdint>
#include <cstddef>

// Problem constants (from reference): N=1, C=D1*D2=1024, H=128, W=160
constexpr int C_  = 1024;
constexpr int H_  = 128;
constexpr int W_  = 160;
constexpr int HW_ = H_ * W_;

typedef uint32_t u32x4 __attribute__((ext_vector_type(4)));
typedef uint32_t u32x8 __attribute__((ext_vector_type(8)));

// ---------------------------------------------------------------------------
// TDM: async-load one contiguous row of 1024 f32 (4 KB) from global into LDS.
// Descriptor per CDNA5 ISA:
//   group0: count=1 | lds_addr | global_addr[56:0] | type=2
//   group1: data_size=4B, tensor_dim0=1024, tensor_dim1=1, tile_dim0=1024,
//           tile_dim1=1, tensor_dim0_stride=1024
// Issued by wave 0 only (TDM ignores EXEC); completion tracked by TENSORcnt.
// ---------------------------------------------------------------------------
__device__ __forceinline__ void tdm_load_row(const float* src, uint32_t lds_byte_off) {
    unsigned long long ga = (unsigned long long)(uintptr_t)src;
    u32x4 g0;
    g0.x = 1u;                                    // count=1, user mode
    g0.y = lds_byte_off;                          // LDS dest (byte offset)
    g0.z = (uint32_t)ga;                          // global_addr lo
    g0.w = (uint32_t)(ga >> 32) | (2u << 30);     // global_addr hi | type=2
    u32x8 g1;
    g1[0] = 2u << 16;                             // data_size = 4 bytes
    g1[1] = 1024u << 16;                          // tensor_dim0 = 1024 (bits 63:48)
    g1[2] = 1u << 16;                             // tensor_dim0 hi=0, tensor_dim1 = 1
    g1[3] = 1024u << 16;                          // tensor_dim1 hi=0, tile_dim0 = 1024
    g1[4] = 1u;                                   // tile_dim1 = 1, tile_dim2 = 0
    g1[5] = 1024u;                                // tensor_dim0_stride = 1024
    g1[6] = 0u;
    g1[7] = 0u;
    asm volatile("tensor_load_to_lds %0, %1" :: "s"(g0), "s"(g1) : "memory");
}

__device__ __forceinline__ uint32_t lds_off_of(const void* p) {
    // Flat pointers to LDS carry the byte offset in the low 32 bits (aperture hi 32).
    return (uint32_t)(uintptr_t)p;
}

// ---------------------------------------------------------------------------
// Transpose x (C,H,W) -> ws (W,H,C): per fixed h, a tiled 1024x160 transpose.
// ---------------------------------------------------------------------------
__global__ void nlf_transpose_chw_whc(const float* __restrict__ in, float* __restrict__ out) {
    __shared__ float tile[32][33];
    const int h  = blockIdx.z;
    const int w0 = blockIdx.x * 32;
    const int c0 = blockIdx.y * 32;
    const int tx = threadIdx.x, ty = threadIdx.y;   // 32 x 8
#pragma unroll
    for (int k = 0; k < 32; k += 8)
        tile[ty + k][tx] = in[(size_t)(c0 + ty + k) * HW_ + (size_t)h * W_ + (w0 + tx)];
    __syncthreads();
#pragma unroll
    for (int k = 0; k < 32; k += 8)
        out[((size_t)(w0 + ty + k) * H_ + h) * C_ + (c0 + tx)] = tile[tx][ty + k];
}

// ---------------------------------------------------------------------------
// One directional scan pass. Block = one line (fixed w for H-scans, fixed h
// for W-scans); 256 threads x 4 channels. Recurrent state (prev row + its
// channel max) lives in LDS. INC: input rows are channel-contiguous -> 3-deep
// TDM ring prefetch. OUTC: output rows channel-contiguous (float4 stores).
// ---------------------------------------------------------------------------
template <bool INC, bool OUTC>
__global__ void nlf_scan(const float* in, float* out, const float* __restrict__ g,
                         int gdir,
                         int in_sl, int in_ss, int in_sc,
                         int out_sl, int out_ss, int out_sc,
                         int S, int g_ss, int g_sl, int rev) {
    __shared__ alignas(16) float prev[C_ + 2];          // halo: prev[0]=prev[C+1]=0
    __shared__ alignas(16) float wmax[8];               // per-wave channel maxes
    __shared__ alignas(16) float stage[INC ? 3 * C_ : 4]; // TDM ring (3 x 4 KB)

    const int tid  = threadIdx.x;
    const int lane = tid & 31;
    const int wid  = tid >> 5;
    const int line = blockIdx.x;
    const int c0   = tid * 4;
    float* prevH = prev + 1;

    if (tid == 0) { prev[0] = 0.0f; prev[C_ + 1] = 0.0f; }

    const float* gbase = g + (size_t)gdir * 5 * HW_;

    if constexpr (INC) {
        if (tid == 0) {                                 // wave 0 issues the ring
#pragma unroll
            for (int r = 0; r < 3; ++r) {
                const int spr = rev ? (S - 1 - r) : r;
                tdm_load_row(in + (long)line * in_sl + (long)spr * in_ss,
                             lds_off_of(&stage[r * C_]));
            }
            __builtin_amdgcn_s_wait_tensorcnt(2);       // row 0 resident
        }
    }
    __syncthreads();

    int slot = 0;                                       // ring slot holding row s
    for (int s = 0; s < S; ++s) {
        const int sp = rev ? (S - 1 - s) : s;

        float x0, x1, x2, x3;
        if constexpr (INC) {
            const float4 xv = *(const float4*)&stage[slot * C_ + c0];
            x0 = xv.x; x1 = xv.y; x2 = xv.z; x3 = xv.w;
        } else {
            const long b = (long)line * in_sl + (long)sp * in_ss;
            x0 = in[b + (long)(c0 + 0) * in_sc];
            x1 = in[b + (long)(c0 + 1) * in_sc];
            x2 = in[b + (long)(c0 + 2) * in_sc];
            x3 = in[b + (long)(c0 + 3) * in_sc];
        }

        float o0, o1, o2, o3;
        if (s == 0) {                                   // boundary: out(0) = x(0)
            o0 = x0; o1 = x1; o2 = x2; o3 = x3;
        } else {
            float pmax = wmax[0];
#pragma unroll
            for (int i = 1; i < 8; ++i) pmax = fmaxf(pmax, wmax[i]);

            // Guidance (uniform per block & step), L1-normalized on the fly.
            const int goff = sp * g_ss + line * g_sl;
            float k0 = gbase[goff];
            float k1 = gbase[goff + HW_];
            float k2 = gbase[goff + 2 * HW_];
            float k3 = gbase[goff + 3 * HW_];
            float k4 = gbase[goff + 4 * HW_];
            float den = fabsf(k0) + fabsf(k1) + fabsf(k2) + fabsf(k3) + fabsf(k4);
            den = fmaxf(den, 1e-12f);
            const float inv = 1.0f / den;
            k0 *= inv; k1 *= inv; k2 *= inv; k3 *= inv; k4 *= inv;

            const float p0 = prevH[c0],     p1 = prevH[c0 + 1];
            const float p2 = prevH[c0 + 2], p3 = prevH[c0 + 3];
            const float pm = prevH[c0 - 1], pp = prevH[c0 + 4];
            o0 = k0 * x0 + k1 * p0 + k2 * pm + k3 * p1 + k4 * pmax;
            o1 = k0 * x1 + k1 * p1 + k2 * p0 + k3 * p2 + k4 * pmax;
            o2 = k0 * x2 + k1 * p2 + k2 * p1 + k3 * p3 + k4 * pmax;
            o3 = k0 * x3 + k1 * p3 + k2 * p2 + k3 * pp + k4 * pmax;
        }

        const long ob = (long)line * out_sl + (long)sp * out_ss;
        if constexpr (OUTC) {
            *(float4*)&out[ob + c0] = make_float4(o0, o1, o2, o3);
        } else {
            out[ob + (long)(c0 + 0) * out_sc] = o0;
            out[ob + (long)(c0 + 1) * out_sc] = o1;
            out[ob + (long)(c0 + 2) * out_sc] = o2;
            out[ob + (long)(c0 + 3) * out_sc] = o3;
        }

        __syncthreads();                                // B1: prev[] / stage(slot) reads done

        prevH[c0] = o0; prevH[c0 + 1] = o1; prevH[c0 + 2] = o2; prevH[c0 + 3] = o3;
        float m = fmaxf(fmaxf(o0, o1), fmaxf(o2, o3));
#pragma unroll
        for (int off = 16; off > 0; off >>= 1)          // wave32 max reduce
            m = fmaxf(m, __shfl_xor(m, off, 32));
        if (lane == 0) wmax[wid] = m;

        if constexpr (INC) {
            if (tid == 0) {
                // <=2 rows outstanding (s+1, s+2); oldest (s+1) must be done.
                __builtin_amdgcn_s_wait_tensorcnt(1);
                if (s + 3 < S) {                        // refill consumed slot with row s+3
                    const int sp3 = rev ? (S - 4 - s) : (s + 3);
                    tdm_load_row(in + (long)line * in_sl + (long)sp3 * in_ss,
                                 lds_off_of(&stage[slot * C_]));
                }
            }
        }
        __syncthreads();                                // B2: publish prev/wmax/stage(s+1)

        slot = (slot == 2) ? 0 : slot + 1;
    }
}

// ---------------------------------------------------------------------------
extern "C" void kernel_launch(void* const* d_in, const int* in_sizes, int n_in,
                              void* d_out, int out_size, void* d_ws, size_t ws_size,
                              hipStream_t stream) {
    const float* x = (const float*)d_in[0];
    const float* g = (const float*)d_in[1];
    float* out = (float*)d_out;
    float* ws  = (float*)d_ws;

    const size_t need = (size_t)C_ * HW_ * sizeof(float);   // 80 MB

    if (ws != nullptr && ws_size >= need) {
        // Primary path: one transpose to (W,H,C); all 4 passes read contiguous
        // channel rows (TDM ring prefetch); final pass writes (C,H,W) directly.
        nlf_transpose_chw_whc<<<dim3(W_ / 32, C_ / 32, H_), dim3(32, 8), 0, stream>>>(x, ws);
        // P1: down (scan h fwd), line = w
        nlf_scan<true, true ><<<W_, 256, 0, stream>>>(ws, ws, g, 0,
            H_ * C_, C_, 1,   H_ * C_, C_, 1,   H_, W_, 1, 0);
        // P2: up (scan h rev)
        nlf_scan<true, true ><<<W_, 256, 0, stream>>>(ws, ws, g, 1,
            H_ * C_, C_, 1,   H_ * C_, C_, 1,   H_, W_, 1, 1);
        // P3: right (scan w fwd), line = h
        nlf_scan<true, true ><<<H_, 256, 0, stream>>>(ws, ws, g, 2,
            C_, H_ * C_, 1,   C_, H_ * C_, 1,   W_, 1, W_, 0);
        // P4: left (scan w rev), write transposed back to (C,H,W) in d_out
        nlf_scan<true, false><<<H_, 256, 0, stream>>>(ws, out, g, 3,
            C_, H_ * C_, 1,   W_, 1, HW_,       W_, 1, W_, 1);
    } else {
        // Fallback: no workspace -> run every pass in the original (C,H,W)
        // layout in-place in d_out (uncoalesced channel gathers, L2-resident).
        nlf_scan<false, false><<<W_, 256, 0, stream>>>(x,   out, g, 0,
            1, W_, HW_,   1, W_, HW_,   H_, W_, 1, 0);
        nlf_scan<false, false><<<W_, 256, 0, stream>>>(out, out, g, 1,
            1, W_, HW_,   1, W_, HW_,   H_, W_, 1, 1);
        nlf_scan<false, false><<<H_, 256, 0, stream>>>(out, out, g, 2,
            W_, 1, HW_,   W_, 1, HW_,   W_, 1, W_, 0);
        nlf_scan<false, false><<<H_, 256, 0, stream>>>(out, out, g, 3,
            W_, 1, HW_,   W_, 1, HW_,   W_, 1, W_, 1);
    }
}